// SpikeFP32LayerNorm_23407571764108
// MI455X (gfx1250) — compile-verified
//
#include <hip/hip_runtime.h>

// ---------------------------------------------------------------------------
// SpikeFP32LayerNorm for MI455X (gfx1250)
// Memory-bound (AI ~ 0.0005 FLOP/B): 256 MiB traffic -> ~11.3 us floor @ 23.3 TB/s.
// Strategy:
//   * double-buffered GLOBAL_LOAD_ASYNC_TO_LDS_B128 row streaming
//     (CDNA5 async path, ASYNCcnt-tracked, th:TH_LOAD_NT - rows are read once)
//   * fp64 sum/sumsq reduction + 5x Newton-Raphson rsqrt (reference numerics)
//   * non-temporal b128 output stores (write-once stream, keep L2 clean)
// ---------------------------------------------------------------------------

typedef float v4f __attribute__((ext_vector_type(4)));

namespace {
constexpr int LN_N        = 4096;   // row length
constexpr int LN_BATCH    = 8192;   // rows
constexpr int LN_THREADS  = 256;    // 8 wave32
constexpr int LN_ROWS_PB  = 4;      // rows per block (double-buffered)
constexpr int LN_GRID     = LN_BATCH / LN_ROWS_PB;        // 2048 blocks
constexpr int LN_CHUNK    = LN_THREADS * 4;               // 1024 floats / round
constexpr int LN_NCHUNK   = LN_N / LN_CHUNK;              // 4 b128 issues / thread
constexpr int LN_WAVES    = LN_THREADS / 32;              // 8
constexpr double LN_EPS   = 1e-6;
constexpr int LN_NR       = 5;
}

// Issue one per-lane async 16B copy: global -> LDS (tracked by ASYNCcnt).
// Non-temporal: each row is consumed exactly once; reuse lives in LDS.
__device__ __forceinline__ void async_ld_b128(unsigned lds_byte_addr, const float* gptr) {
    unsigned long long ga = (unsigned long long)(size_t)gptr;
    asm volatile("global_load_async_to_lds_b128 %0, %1, off th:TH_LOAD_NT"
                 :: "v"(lds_byte_addr), "v"(ga)
                 : "memory");
}

__device__ __forceinline__ void wait_asynccnt_le4() {
    asm volatile("s_wait_asynccnt 0x4" ::: "memory");
}
__device__ __forceinline__ void wait_asynccnt_0() {
    asm volatile("s_wait_asynccnt 0x0" ::: "memory");
}

__global__ __launch_bounds__(LN_THREADS)
void spike_fp32_layernorm_kernel(const float* __restrict__ x, float* __restrict__ out) {
    __shared__ __align__(16) float buf[2][LN_N];    // 32 KB double buffer
    __shared__ double red_s[LN_WAVES];
    __shared__ double red_q[LN_WAVES];
    __shared__ double bcast[2];                     // mean, y

    const int tid  = threadIdx.x;
    const int wave = tid >> 5;
    const int lane = tid & 31;
    const int row0 = blockIdx.x;

    // ---- prefetch first row into buf[0] (4 async b128 per thread) ----
    {
        const float* src = x + (size_t)row0 * LN_N;
        #pragma unroll
        for (int j = 0; j < LN_NCHUNK; ++j) {
            const int off = j * LN_CHUNK + tid * 4;
            const unsigned lds = (unsigned)(size_t)(const void*)&buf[0][off];
            async_ld_b128(lds, src + off);
        }
    }

    for (int i = 0; i < LN_ROWS_PB; ++i) {
        const int cur = i & 1;
        const int row = row0 + i * LN_GRID;

        // ---- prefetch next row into the other buffer, then wait for current ----
        if (i + 1 < LN_ROWS_PB) {
            const float* src = x + (size_t)(row + LN_GRID) * LN_N;
            const int nb = cur ^ 1;
            #pragma unroll
            for (int j = 0; j < LN_NCHUNK; ++j) {
                const int off = j * LN_CHUNK + tid * 4;
                const unsigned lds = (unsigned)(size_t)(const void*)&buf[nb][off];
                async_ld_b128(lds, src + off);
            }
            wait_asynccnt_le4();   // async loads retire in order: current buffer done
        } else {
            wait_asynccnt_0();
        }
        __syncthreads();

        // ---- pass 1: fp64 sum and sum-of-squares over the row ----
        double s = 0.0, q = 0.0;
        #pragma unroll
        for (int j = 0; j < LN_NCHUNK; ++j) {
            const int off = j * LN_CHUNK + tid * 4;
            const v4f v = *(const v4f*)&buf[cur][off];
            const double d0 = (double)v.x, d1 = (double)v.y;
            const double d2 = (double)v.z, d3 = (double)v.w;
            s += d0; s += d1; s += d2; s += d3;
            q = fma(d0, d0, q); q = fma(d1, d1, q);
            q = fma(d2, d2, q); q = fma(d3, d3, q);
        }
        // wave32 butterfly reduction
        #pragma unroll
        for (int o = 16; o > 0; o >>= 1) {
            s += __shfl_xor(s, o);
            q += __shfl_xor(q, o);
        }
        if (lane == 0) { red_s[wave] = s; red_q[wave] = q; }
        __syncthreads();

        if (tid == 0) {
            double ts = 0.0, tq = 0.0;
            #pragma unroll
            for (int w = 0; w < LN_WAVES; ++w) { ts += red_s[w]; tq += red_q[w]; }
            const double invN = 1.0 / (double)LN_N;        // exact power of two
            const double mean = ts * invN;
            const double var  = fma(-mean, mean, tq * invN);
            const double vpe  = var + LN_EPS;
            double y = 1.0 / vpe;                          // fp64 divide seed
            #pragma unroll
            for (int it = 0; it < LN_NR; ++it)             // Newton-Raphson, as reference
                y = 0.5 * y * (3.0 - vpe * (y * y));
            bcast[0] = mean;
            bcast[1] = y;
        }
        __syncthreads();

        const double mean = bcast[0];
        const double yr   = bcast[1];
        const double nmy  = -mean * yr;

        // ---- pass 2: out = (x - mean) * y, fp32 narrow, NT b128 stores ----
        float* dst = out + (size_t)row * LN_N;
        #pragma unroll
        for (int j = 0; j < LN_NCHUNK; ++j) {
            const int off = j * LN_CHUNK + tid * 4;
            const v4f v = *(const v4f*)&buf[cur][off];
            v4f o4;
            o4.x = (float)fma((double)v.x, yr, nmy);
            o4.y = (float)fma((double)v.y, yr, nmy);
            o4.z = (float)fma((double)v.z, yr, nmy);
            o4.w = (float)fma((double)v.w, yr, nmy);
            __builtin_nontemporal_store(o4, (v4f*)&dst[off]);
        }
        __syncthreads();   // all reads of buf[cur] done before next iter refills it
    }
}

extern "C" void kernel_launch(void* const* d_in, const int* in_sizes, int n_in,
                              void* d_out, int out_size, void* d_ws, size_t ws_size,
                              hipStream_t stream) {
    (void)in_sizes; (void)n_in; (void)d_ws; (void)ws_size; (void)out_size;
    const float* x  = (const float*)d_in[0];
    float* out      = (float*)d_out;
    spike_fp32_layernorm_kernel<<<LN_GRID, LN_THREADS, 0, stream>>>(x, out);
}